// GraphAttentionLayer_2199023255708
// MI455X (gfx1250) — compile-verified
//
#include <hip/hip_runtime.h>
#include <hip/hip_bf16.h>

typedef float v2f __attribute__((ext_vector_type(2)));
typedef float v8f __attribute__((ext_vector_type(8)));

#define GAT_N 8192
#define GAT_IN 512
#define GAT_OUT 64
#define GAT_ALPHA 0.2f
#define GAT_NEGFILL -9e15f

// ---------------------------------------------------------------------------
// Kernel 1: h = X @ W   (8192x512 @ 512x64), f32 WMMA 16x16x4.
// One wave per 16x16 output tile; 2048 waves, 8 waves/block.
// A layout (16x4 f32): lane<16 -> M=lane, {v0,v1}={K0,K1}; lane>=16 -> {K2,K3}
// B layout (4x16 f32): lane%16 -> N,    half0 {v0,v1}={K0,K1}; half1 {K2,K3}
// C layout: vgpr r, lane l -> row r+8*(l>=16), col l%16
// ---------------------------------------------------------------------------
__global__ __launch_bounds__(256) void gat_hgemm(const float* __restrict__ X,
                                                 const float* __restrict__ W,
                                                 float* __restrict__ h) {
  const int wave = (blockIdx.x * blockDim.x + threadIdx.x) >> 5;
  const int lane = threadIdx.x & 31;
  const int half = lane >> 4;
  const int lr   = lane & 15;
  const int mt = wave >> 2;   // 512 M tiles
  const int nt = wave & 3;    // 4 N tiles
  const int m0 = mt * 16, n0 = nt * 16;

  v8f c = {};
  const float* xrow = X + (size_t)(m0 + lr) * GAT_IN + 2 * half;
  const float* wcol = W + (size_t)(2 * half) * GAT_OUT + n0 + lr;

  for (int k0 = 0; k0 < GAT_IN; k0 += 4) {
    v2f a = *(const v2f*)(xrow + k0);
    v2f b;
    b.x = wcol[k0 * GAT_OUT];
    b.y = wcol[k0 * GAT_OUT + GAT_OUT];
    c = __builtin_amdgcn_wmma_f32_16x16x4_f32(false, a, false, b, (short)0, c,
                                              false, false);
  }

  float* o = h + (size_t)(m0 + 8 * half) * GAT_OUT + n0 + lr;
#pragma unroll
  for (int r = 0; r < 8; ++r) o[r * GAT_OUT] = c[r];
}

// ---------------------------------------------------------------------------
// Kernel 2: h1 = h @ a[:64], h2 = h @ a[64:]. One thread per row.
// ---------------------------------------------------------------------------
__global__ __launch_bounds__(256) void gat_proj(const float* __restrict__ h,
                                                const float* __restrict__ a,
                                                float* __restrict__ h1,
                                                float* __restrict__ h2) {
  const int i = blockIdx.x * blockDim.x + threadIdx.x;
  if (i >= GAT_N) return;
  const float* hr = h + (size_t)i * GAT_OUT;
  float s1 = 0.f, s2 = 0.f;
#pragma unroll 16
  for (int k = 0; k < GAT_OUT; ++k) {
    float v = hr[k];
    s1 = fmaf(v, a[k], s1);
    s2 = fmaf(v, a[GAT_OUT + k], s2);
  }
  h1[i] = s1;
  h2[i] = s2;
}

// ---------------------------------------------------------------------------
// Kernel 3: fused flash-softmax attention + attn@h + ELU.
// 512 blocks x 16 rows; 8 waves/block; wave w owns j-chunks w, w+8, ...
// (16 columns per chunk). Online (m, l, acc) per wave, split-softmax merge
// across waves in LDS at the end.
// ---------------------------------------------------------------------------
__global__ __launch_bounds__(256) void gat_attn(const int* __restrict__ adj,
                                                const float* __restrict__ h,
                                                const float* __restrict__ h1,
                                                const float* __restrict__ h2,
                                                float* __restrict__ out) {
  __shared__ float sm[8][16];
  __shared__ float ss[8][16];
  __shared__ float sacc[8][16][GAT_OUT];  // 32 KB

  const int i0   = blockIdx.x * 16;
  const int wave = threadIdx.x >> 5;
  const int lane = threadIdx.x & 31;
  const int half = lane >> 4;
  const int lr   = lane & 15;
  const int row  = i0 + lr;  // attention row this lane scores

  const float my_h1 = h1[row];
  const int* adjrow = adj + (size_t)row * GAT_N;

  float mrow = -INFINITY;  // running row max (row = lane%16)
  float srow = 0.f;        // running row sum of exp
  float m8[8];             // running max for acc rows r+8*half
#pragma unroll
  for (int r = 0; r < 8; ++r) m8[r] = -INFINITY;
  v8f acc0 = {}, acc1 = {}, acc2 = {}, acc3 = {};

  for (int c = wave; c < GAT_N / 16; c += 8) {
    const int j0 = c * 16;

    // --- scores for this lane's 8 (row, j) pairs, masked + LeakyReLU ---
    float e[8];
    float cmax = -INFINITY;
#pragma unroll
    for (int t = 0; t < 4; ++t) {
      const int j = j0 + 4 * t + 2 * half;
      int2 ad = *(const int2*)(adjrow + j);
      float2 hh = *(const float2*)(h2 + j);
      float e0 = my_h1 + hh.x;
      float e1 = my_h1 + hh.y;
      e0 = e0 > 0.f ? e0 : GAT_ALPHA * e0;
      e1 = e1 > 0.f ? e1 : GAT_ALPHA * e1;
      e0 = ad.x > 0 ? e0 : GAT_NEGFILL;
      e1 = ad.y > 0 ? e1 : GAT_NEGFILL;
      e[2 * t] = e0;
      e[2 * t + 1] = e1;
      cmax = fmaxf(cmax, fmaxf(e0, e1));
    }
    // per-row chunk max: reduce over lane pair (i, i+16)
    cmax = fmaxf(cmax, __shfl_xor(cmax, 16));

    const bool needRescale = (__ballot(cmax > mrow) != 0ull);  // wave-uniform
    const float mnew = fmaxf(mrow, cmax);
    if (needRescale) {
#pragma unroll
      for (int r = 0; r < 8; ++r) {
        // running max of acc row r+8*half lives in lane (r+8*half)
        float mn = __shfl(mnew, r + 8 * half);
        float sc = __expf(m8[r] - mn);
        m8[r] = mn;
        acc0[r] *= sc;
        acc1[r] *= sc;
        acc2[r] *= sc;
        acc3[r] *= sc;
      }
      srow *= __expf(mrow - mnew);
      mrow = mnew;
    }

    // --- p = exp(e - m), row-sum update ---
    float p[8];
    float psum = 0.f;
#pragma unroll
    for (int q = 0; q < 8; ++q) {
      p[q] = __expf(e[q] - mrow);
      psum += p[q];
    }
    psum += __shfl_xor(psum, 16);
    srow += psum;

    // --- acc += P(16xK) @ h(Kx64): 4 K-slices x 4 N-tiles of f32 WMMA ---
#pragma unroll
    for (int t = 0; t < 4; ++t) {
      v2f a;
      a.x = p[2 * t];
      a.y = p[2 * t + 1];
      const float* hb = h + (size_t)(j0 + 4 * t + 2 * half) * GAT_OUT + lr;
      v2f b0, b1, b2, b3;
      b0.x = hb[0];  b0.y = hb[GAT_OUT + 0];
      b1.x = hb[16]; b1.y = hb[GAT_OUT + 16];
      b2.x = hb[32]; b2.y = hb[GAT_OUT + 32];
      b3.x = hb[48]; b3.y = hb[GAT_OUT + 48];
      acc0 = __builtin_amdgcn_wmma_f32_16x16x4_f32(false, a, false, b0, (short)0, acc0, false, false);
      acc1 = __builtin_amdgcn_wmma_f32_16x16x4_f32(false, a, false, b1, (short)0, acc1, false, false);
      acc2 = __builtin_amdgcn_wmma_f32_16x16x4_f32(false, a, false, b2, (short)0, acc2, false, false);
      acc3 = __builtin_amdgcn_wmma_f32_16x16x4_f32(false, a, false, b3, (short)0, acc3, false, false);
    }
  }

  // --- publish per-wave partials ---
  if (half == 0) {
    sm[wave][lr] = mrow;
    ss[wave][lr] = srow;
  }
#pragma unroll
  for (int r = 0; r < 8; ++r) {
    sacc[wave][r + 8 * half][lr +  0] = acc0[r];
    sacc[wave][r + 8 * half][lr + 16] = acc1[r];
    sacc[wave][r + 8 * half][lr + 32] = acc2[r];
    sacc[wave][r + 8 * half][lr + 48] = acc3[r];
  }
  __syncthreads();

  // --- split-softmax merge of the 8 wave partials, then ELU + store ---
  const int col   = threadIdx.x & 63;
  const int rbase = threadIdx.x >> 6;  // 0..3
#pragma unroll
  for (int rr = 0; rr < 4; ++rr) {
    const int i = rbase + rr * 4;
    float M = -INFINITY;
#pragma unroll
    for (int w = 0; w < 8; ++w) M = fmaxf(M, sm[w][i]);
    float S = 0.f, A = 0.f;
#pragma unroll
    for (int w = 0; w < 8; ++w) {
      float sc = __expf(sm[w][i] - M);
      S += ss[w][i] * sc;
      A += sacc[w][i][col] * sc;
    }
    float o = A / S;
    o = o > 0.f ? o : (__expf(o) - 1.f);  // ELU (alpha=1)
    out[(size_t)(i0 + i) * GAT_OUT + col] = o;
  }
}

// ---------------------------------------------------------------------------
// Host launcher
// ---------------------------------------------------------------------------
extern "C" void kernel_launch(void* const* d_in, const int* in_sizes, int n_in,
                              void* d_out, int out_size, void* d_ws, size_t ws_size,
                              hipStream_t stream) {
  const float* features = (const float*)d_in[0];  // [8192, 512]
  const int*   adj      = (const int*)d_in[1];    // [8192, 8192]
  const float* W        = (const float*)d_in[2];  // [512, 64]
  const float* a        = (const float*)d_in[3];  // [128, 1]
  float* out = (float*)d_out;                     // [8192, 64]

  float* h  = (float*)d_ws;                       // 8192*64 f32 = 2 MB
  float* h1 = h + (size_t)GAT_N * GAT_OUT;        // 32 KB
  float* h2 = h1 + GAT_N;                         // 32 KB

  // 2048 tile-waves, 8 waves per 256-thread block
  gat_hgemm<<<256, 256, 0, stream>>>(features, W, h);
  gat_proj<<<GAT_N / 256, 256, 0, stream>>>(h, a, h1, h2);
  gat_attn<<<GAT_N / 16, 256, 0, stream>>>(adj, h, h1, h2, out);
}